// GCN_64098091925532
// MI455X (gfx1250) — compile-verified
//
#include <hip/hip_runtime.h>
#include <hip/hip_bf16.h>

// GCN forward for MI455X (gfx1250, wave32).
// Dense per-node 10x10 transforms run on V_WMMA_F32_16X16X4_F32 (fp32 WMMA,
// one 16-node tile per wave, K padded 10->12, N padded 10->16).
// All fragment builds are branchless (clamped-index loads + v_cndmask selects)
// so EXEC stays all-ones and no saveexec forests appear.
// Edge scatter uses coalesced float4 gathers + global f32 atomics.

#define H  10   // feature width
#define HP 16   // padded row stride (floats)

typedef float v2f __attribute__((ext_vector_type(2)));
typedef float v8f __attribute__((ext_vector_type(8)));

__device__ __forceinline__ float relu_f(float x) { return fmaxf(x, 0.0f); }

// B fragment for 16x16x4 f32 WMMA from row-major [H x H] weight, zero-padded
// to 12x16. Loads are unconditional (clamped indices); masking is via selects.
__device__ __forceinline__ void load_b_frags(const float* __restrict__ W, v2f bf[3]) {
    const int lane = threadIdx.x & 31;
    const int col  = lane & 15;
    const int colc = (col < H) ? col : 0;
    const int koff = (lane >> 4) << 1;   // 0 or 2
#pragma unroll
    for (int s = 0; s < 3; ++s) {
        const int k0  = 4 * s + koff;
        const int k1  = k0 + 1;
        const int kc0 = (k0 < H) ? k0 : 0;
        const int kc1 = (k1 < H) ? k1 : 0;
        const float w0 = W[kc0 * H + colc];   // unconditional load
        const float w1 = W[kc1 * H + colc];   // unconditional load
        v2f t;
        t.x = (k0 < H && col < H) ? w0 : 0.0f;
        t.y = (k1 < H && col < H) ? w1 : 0.0f;
        bf[s] = t;
    }
}

__global__ void k_zero(float* __restrict__ p, size_t n) {
    size_t i = (size_t)blockIdx.x * blockDim.x + threadIdx.x;
    const size_t stride = (size_t)gridDim.x * blockDim.x;
    for (; i < n; i += stride) p[i] = 0.0f;
}

__global__ void k_degrees(const int* __restrict__ src, const int* __restrict__ dst,
                          float* __restrict__ degv, float* __restrict__ degc, int e) {
    int i = blockIdx.x * blockDim.x + threadIdx.x;
    if (i >= e) return;
    atomicAdd(&degv[src[i]], 1.0f);
    atomicAdd(&degc[dst[i]], 1.0f);
}

// hv[row] = relu([c,x]@Wv + bv) @ W2 * deg_v(row)^-1/2     (src-norm fused)
__global__ void k_var_embed(const float* __restrict__ var_c, const float* __restrict__ var_x,
                            const float* __restrict__ Wv, const float* __restrict__ bv,
                            const float* __restrict__ W2, const float* __restrict__ degv,
                            float* __restrict__ hv, int nv) {
    const int lane = threadIdx.x & 31;
    const int wave = threadIdx.x >> 5;
    const int tile = blockIdx.x * (blockDim.x >> 5) + wave;
    const int ntiles = (nv + 15) >> 4;
    if (tile >= ntiles) return;
    const int base = tile << 4;
    const int r    = lane & 15;
    const int row  = base + r;
    const int rowc = (row < nv) ? row : 0;
    const bool rowok = (row < nv);
    const int koff = (lane >> 4) << 1;

    v2f bf[3];
    load_b_frags(W2, bf);

    const float c0 = var_c[rowc];
    const float x0 = var_x[rowc];

    v8f acc = {};
#pragma unroll
    for (int s = 0; s < 3; ++s) {
        const int k0  = 4 * s + koff, k1 = k0 + 1;
        const int kc0 = (k0 < H) ? k0 : 0;
        const int kc1 = (k1 < H) ? k1 : 0;
        // unconditional clamped loads, then select
        const float e0 = relu_f(fmaf(c0, Wv[kc0], fmaf(x0, Wv[H + kc0], bv[kc0])));
        const float e1 = relu_f(fmaf(c0, Wv[kc1], fmaf(x0, Wv[H + kc1], bv[kc1])));
        v2f a;
        a.x = (rowok && k0 < H) ? e0 : 0.0f;
        a.y = (rowok && k1 < H) ? e1 : 0.0f;
        acc = __builtin_amdgcn_wmma_f32_16x16x4_f32(false, a, false, bf[s],
                                                    (short)0, acc, false, false);
    }

    const int col = lane & 15;
    const int mh  = (lane >> 4) << 3;   // 0 or 8
    if (base + 16 <= nv) {              // wave-uniform fast path (always for NV%16==0)
#pragma unroll
        for (int j = 0; j < 8; ++j) {
            const int m = base + mh + j;
            const float nrm = rsqrtf(fmaxf(degv[m], 1.0f));
            hv[(size_t)m * HP + col] = acc[j] * nrm;
        }
    } else {
#pragma unroll
        for (int j = 0; j < 8; ++j) {
            const int m = base + mh + j;
            if (m < nv) {
                const float nrm = rsqrtf(fmaxf(degv[m], 1.0f));
                hv[(size_t)m * HP + col] = acc[j] * nrm;
            }
        }
    }
}

// agg[d] += h[s] * A[e]  (10 f32 atomics per edge; rows are 64B-aligned)
__global__ void k_scatter(const int* __restrict__ sidx, const int* __restrict__ didx,
                          const float* __restrict__ ew, const float* __restrict__ hsrc,
                          float* __restrict__ aggd, int e) {
    int i = blockIdx.x * blockDim.x + threadIdx.x;
    if (i >= e) return;
    const int s = sidx[i];
    const int d = didx[i];
    const float a = ew[i];
    const float4* h4 = (const float4*)(hsrc + (size_t)s * HP);
    const float4 v0 = h4[0], v1 = h4[1], v2 = h4[2];
    float* ar = aggd + (size_t)d * HP;
    atomicAdd(&ar[0], v0.x * a); atomicAdd(&ar[1], v0.y * a);
    atomicAdd(&ar[2], v0.z * a); atomicAdd(&ar[3], v0.w * a);
    atomicAdd(&ar[4], v1.x * a); atomicAdd(&ar[5], v1.y * a);
    atomicAdd(&ar[6], v1.z * a); atomicAdd(&ar[7], v1.w * a);
    atomicAdd(&ar[8], v2.x * a); atomicAdd(&ar[9], v2.y * a);
}

// Branchless A fragment from a zero-padded [*,HP] activation buffer:
//   a[k] = relu(agg[row][k] * nrm + b2[k])  for k<H, exact 0 in pads.
// Loads the (k0,k1) pair as one aligned 8-byte v2f (pads are zeroed).
__device__ __forceinline__ v2f act_pair(const float* __restrict__ ar, float nrm,
                                        const float* __restrict__ b2,
                                        int k0, bool rowok) {
    const v2f av = *(const v2f*)(ar + k0);               // k0 even, <=10 -> in padded row
    const int kc0 = (k0 < H) ? k0 : 0;
    const int kc1 = (k0 + 1 < H) ? k0 + 1 : 0;
    const float b0 = b2[kc0];
    const float b1 = b2[kc1];
    v2f a;
    a.x = (rowok && k0 < H)     ? relu_f(fmaf(av.x, nrm, b0)) : 0.0f;
    a.y = (rowok && k0 + 1 < H) ? relu_f(fmaf(av.y, nrm, b1)) : 0.0f;
    return a;
}

// hc[row] = relu(aggc*norm_c + b2) @ W2 * deg_c(row)^-1/2
__global__ void k_con_update(const float* __restrict__ aggc, const float* __restrict__ b2,
                             const float* __restrict__ W2, const float* __restrict__ degc,
                             float* __restrict__ hc, int nc) {
    const int lane = threadIdx.x & 31;
    const int wave = threadIdx.x >> 5;
    const int tile = blockIdx.x * (blockDim.x >> 5) + wave;
    const int ntiles = (nc + 15) >> 4;
    if (tile >= ntiles) return;
    const int base = tile << 4;
    const int r    = lane & 15;
    const int row  = base + r;
    const int rowc = (row < nc) ? row : 0;
    const bool rowok = (row < nc);
    const int koff = (lane >> 4) << 1;

    v2f bf[3];
    load_b_frags(W2, bf);

    const float nrm_in = rsqrtf(fmaxf(degc[rowc], 1.0f));
    const float* ar = aggc + (size_t)rowc * HP;

    v8f acc = {};
#pragma unroll
    for (int s = 0; s < 3; ++s) {
        const v2f a = act_pair(ar, nrm_in, b2, 4 * s + koff, rowok);
        acc = __builtin_amdgcn_wmma_f32_16x16x4_f32(false, a, false, bf[s],
                                                    (short)0, acc, false, false);
    }

    const int col = lane & 15;
    const int mh  = (lane >> 4) << 3;
    if (base + 16 <= nc) {
#pragma unroll
        for (int j = 0; j < 8; ++j) {
            const int m = base + mh + j;
            const float nrm = rsqrtf(fmaxf(degc[m], 1.0f));
            hc[(size_t)m * HP + col] = acc[j] * nrm;
        }
    } else {
#pragma unroll
        for (int j = 0; j < 8; ++j) {
            const int m = base + mh + j;
            if (m < nc) {
                const float nrm = rsqrtf(fmaxf(degc[m], 1.0f));
                hc[(size_t)m * HP + col] = acc[j] * nrm;
            }
        }
    }
}

// h_var = relu(aggv*norm_v + b2); MLP Wo1->Wo2->Wo3; accumulate sum of logits.
__global__ void k_var_out(const float* __restrict__ aggv, const float* __restrict__ degv,
                          const float* __restrict__ b2,
                          const float* __restrict__ Wo1, const float* __restrict__ bo1,
                          const float* __restrict__ Wo2, const float* __restrict__ bo2,
                          const float* __restrict__ Wo3,
                          float* __restrict__ acc_out, int nv) {
    __shared__ float lds[8][16 * 16];
    const int lane = threadIdx.x & 31;
    const int wave = threadIdx.x >> 5;
    const int tile = blockIdx.x * (blockDim.x >> 5) + wave;
    const int ntiles = (nv + 15) >> 4;
    if (tile >= ntiles) return;
    const int base = tile << 4;
    const int r    = lane & 15;
    const int row  = base + r;
    const int rowc = (row < nv) ? row : 0;
    const bool rowok = (row < nv);
    const int koff = (lane >> 4) << 1;
    const int col  = lane & 15;
    const int colc = (col < H) ? col : 0;
    const int mh   = (lane >> 4) << 3;

    v2f bf1[3], bf2[3];
    load_b_frags(Wo1, bf1);
    load_b_frags(Wo2, bf2);

    const float nrm_in = rsqrtf(fmaxf(degv[rowc], 1.0f));
    const float* ar = aggv + (size_t)rowc * HP;

    // h_var @ Wo1
    v8f c1 = {};
#pragma unroll
    for (int s = 0; s < 3; ++s) {
        const v2f a = act_pair(ar, nrm_in, b2, 4 * s + koff, rowok);
        c1 = __builtin_amdgcn_wmma_f32_16x16x4_f32(false, a, false, bf1[s],
                                                   (short)0, c1, false, false);
    }

    // h1 = relu(c1 + bo1) -> LDS (transpose C-layout back to A-layout)
    const float bo1c = bo1[colc];
#pragma unroll
    for (int j = 0; j < 8; ++j) {
        const float v = (col < H) ? relu_f(c1[j] + bo1c) : 0.0f;
        lds[wave][(mh + j) * 16 + col] = v;
    }
    asm volatile("s_wait_dscnt 0" ::: "memory");   // wave-local LDS ordering

    // h1 @ Wo2 (pads in LDS are exact zeros -> unconditional v2f reads)
    v8f c2 = {};
#pragma unroll
    for (int s = 0; s < 3; ++s) {
        const int k0 = 4 * s + koff;
        const v2f a = *(const v2f*)&lds[wave][r * 16 + k0];
        c2 = __builtin_amdgcn_wmma_f32_16x16x4_f32(false, a, false, bf2[s],
                                                   (short)0, c2, false, false);
    }

    // logit contribution: sum over rows & cols of relu(c2 + bo2) * Wo3[col]
    const float w3  = (col < H) ? Wo3[colc] : 0.0f;
    const float bb2 = bo2[colc];
    float t = 0.0f;
    if (base + 16 <= nv) {
#pragma unroll
        for (int j = 0; j < 8; ++j) t += relu_f(c2[j] + bb2) * w3;
    } else {
#pragma unroll
        for (int j = 0; j < 8; ++j) {
            const int m = base + mh + j;
            const float h2 = relu_f(c2[j] + bb2) * w3;
            t += (m < nv) ? h2 : 0.0f;
        }
    }
#pragma unroll
    for (int off = 16; off > 0; off >>= 1) t += __shfl_xor(t, off, 32);
    if (lane == 0) atomicAdd(acc_out, t);
}

__global__ void k_finalize(const float* __restrict__ acc, const float* __restrict__ bo3,
                           float* __restrict__ out, float inv_n) {
    out[0] = acc[0] * inv_n + bo3[0];
}

extern "C" void kernel_launch(void* const* d_in, const int* in_sizes, int n_in,
                              void* d_out, int out_size, void* d_ws, size_t ws_size,
                              hipStream_t stream) {
    const float* var_c = (const float*)d_in[0];
    const float* var_x = (const float*)d_in[1];
    // d_in[2] con_b unused on the live path (GraphConv ignores dst feats)
    const float* A   = (const float*)d_in[3];
    const int*   src = (const int*)d_in[4];
    const int*   dst = (const int*)d_in[5];
    const float* Wv  = (const float*)d_in[6];
    const float* bv  = (const float*)d_in[7];
    // d_in[8..11] Wc,bc,W1,b1 dead (conv1 output overwritten)
    const float* W2  = (const float*)d_in[12];
    const float* b2  = (const float*)d_in[13];
    const float* Wo1 = (const float*)d_in[14];
    const float* bo1 = (const float*)d_in[15];
    const float* Wo2 = (const float*)d_in[16];
    const float* bo2 = (const float*)d_in[17];
    const float* Wo3 = (const float*)d_in[18];
    const float* bo3 = (const float*)d_in[19];

    const int nv = in_sizes[0];
    const int nc = in_sizes[2];
    const int e  = in_sizes[3];

    // workspace layout (floats); zeroed region first, then hv/hc (fully written)
    float* ws = (float*)d_ws;
    size_t o = 0;
    float* degv = ws + o; o += (size_t)nv;
    float* degc = ws + o; o += (size_t)nc;
    float* aggc = ws + o; o += (size_t)nc * HP;
    float* aggv = ws + o; o += (size_t)nv * HP;
    float* acc  = ws + o; o += 4;                 // pad to keep 16B alignment
    const size_t zcount = o;
    float* hv   = ws + o; o += (size_t)nv * HP;
    float* hc   = ws + o; o += (size_t)nc * HP;
    (void)ws_size; (void)n_in; (void)out_size;

    const int tb = 256;
    const int eblocks = (e + tb - 1) / tb;
    const int vtiles = (nv + 15) / 16;
    const int ctiles = (nc + 15) / 16;

    k_zero<<<2048, tb, 0, stream>>>(ws, zcount);
    k_degrees<<<eblocks, tb, 0, stream>>>(src, dst, degv, degc, e);
    k_var_embed<<<(vtiles + 7) / 8, tb, 0, stream>>>(var_c, var_x, Wv, bv, W2, degv, hv, nv);
    k_scatter<<<eblocks, tb, 0, stream>>>(src, dst, A, hv, aggc, e);
    k_con_update<<<(ctiles + 7) / 8, tb, 0, stream>>>(aggc, b2, W2, degc, hc, nc);
    k_scatter<<<eblocks, tb, 0, stream>>>(dst, src, A, hc, aggv, e);
    k_var_out<<<(vtiles + 7) / 8, tb, 0, stream>>>(aggv, degv, b2, Wo1, bo1, Wo2, bo2, Wo3,
                                                   acc, nv);
    k_finalize<<<1, 1, 0, stream>>>(acc, bo3, (float*)d_out, 1.0f / (float)nv);
}